// SparkNet_19997367730511
// MI455X (gfx1250) — compile-verified
//
#include <hip/hip_runtime.h>
#include <hip/hip_bf16.h>
#include <math.h>

#define NN      8192
#define KS      256
#define GRIDW   128
#define KSPLIT  4
#define KCHUNK  (NN / KSPLIT)   // 2048

typedef __attribute__((ext_vector_type(2))) float    v2f;
typedef __attribute__((ext_vector_type(8))) float    v8f;
typedef __attribute__((ext_vector_type(4))) unsigned v4u;
typedef __attribute__((ext_vector_type(8))) int      v8i;
typedef __attribute__((ext_vector_type(4))) int      v4i;

__device__ __forceinline__ unsigned pcg_hash(unsigned x) {
    x = x * 747796405u + 2891336453u;
    unsigned w = ((x >> ((x >> 28u) + 4u)) ^ x) * 277803737u;
    return (w >> 22u) ^ w;
}
__device__ __forceinline__ float u01(unsigned h) {
    return (float)(h >> 8) * 0x1.0p-24f + 0x1.0p-25f;
}
__device__ __forceinline__ float clip1(float v) {
    return fminf(1.0f, fmaxf(-1.0f, v));
}

// ---------------------------------------------------------------------------
// Kernel A (fused): single streaming pass over W.
//  * accumulates y_part[by] = W[:, kchunk] @ s[kchunk] via V_WMMA_F32_16X16X4_F32
//  * simultaneously emits Wout = clip(0.999*W) for every element it loads
//    (the bx/by/lane/kk tiling covers W exactly once), saving a second
//    268 MB read pass at 23.3 TB/s.
//  * the s chunk is staged into LDS by the Tensor Data Mover (1-D D#,
//    issued from wave 0; TENSORcnt wait + barrier before consumption).
// ---------------------------------------------------------------------------
__global__ __launch_bounds__(256) void matvec_fused(
    const float* __restrict__ W, const float* __restrict__ s,
    float* __restrict__ Wout, float* __restrict__ partial /* KSPLIT*NN in ws */) {
    __shared__ float sBuf[KCHUNK];
    const int kBase = blockIdx.y * KCHUNK;

    if ((threadIdx.x >> 5) == 0) {   // wave 0 issues the TDM (EXEC ignored by TDM)
        const unsigned ldsOff = (unsigned)(size_t)&sBuf[0];      // LDS byte offset
        const unsigned long long gaddr = (unsigned long long)(size_t)(s + kBase);
        v4u g0;
        g0.x = 1u;                                               // count=1, user D#
        g0.y = ldsOff;                                           // lds_addr
        g0.z = (unsigned)(gaddr & 0xFFFFFFFFull);                // global_addr[31:0]
        g0.w = (unsigned)((gaddr >> 32) & 0x1FFFFFFull) | (2u << 30); // addr[56:32] | type=2
        v8i g1;
        g1[0] = 0x20000;                 // workgroup_mask=0, data_size=2 (4 bytes)
        g1[1] = (int)(KCHUNK << 16);     // tensor_dim0 = 2048 (low 16 in [31:16])
        g1[2] = 0x00010000;              // tensor_dim0 hi=0, tensor_dim1 = 1
        g1[3] = (int)(KCHUNK << 16);     // tensor_dim1 hi=0, tile_dim0 = 2048
        g1[4] = 1;                       // tile_dim1 = 1, tile_dim2 = 0
        g1[5] = KCHUNK;                  // tensor_dim0_stride[31:0]
        g1[6] = 0;
        g1[7] = 0;
        v4i z4 = {0, 0, 0, 0};
        v8i z8 = {0, 0, 0, 0, 0, 0, 0, 0};
        __builtin_amdgcn_tensor_load_to_lds(g0, g1, z4, z4, z8, 0);
        __builtin_amdgcn_s_wait_tensorcnt(0);
    }
    __syncthreads();

    const int wid  = threadIdx.x >> 5;
    const int lane = threadIdx.x & 31;
    const int m    = lane & 15;   // A-matrix row (lanes 0-15 / 16-31 both map M=0..15)
    const int hb   = lane >> 4;   // K phase: {0,1} vs {2,3} per ISA 16x4 f32 A layout
    const int rowBase = blockIdx.x * 128 + wid * 16;
    const size_t off = (size_t)(rowBase + m) * NN + kBase + hb * 2;
    const float* rowPtr = W + off;
    float*       outPtr = Wout + off;

    v8f acc = {};
    #pragma unroll 4
    for (int kk = 0; kk < KCHUNK; kk += 4) {
        v2f a = *(const v2f*)(rowPtr + kk);   // global_load_b64
        v2f b;                                // ds_load from TDM-staged chunk
        b.x = sBuf[kk + hb * 2];
        b.y = sBuf[kk + hb * 2 + 1];
        acc = __builtin_amdgcn_wmma_f32_16x16x4_f32(
            false, a, false, b, (short)0, acc, false, false);
        v2f w;                                // fused decay+clip write-back
        w.x = clip1(a.x * 0.999f);
        w.y = clip1(a.y * 0.999f);
        __builtin_nontemporal_store(w, (v2f*)(outPtr + kk));
    }

    // C/D layout: lane 0 holds M=0..7 (N=0) in acc[0..7]; lane 16 holds M=8..15.
    float* out = partial + (size_t)blockIdx.y * NN + rowBase;
    if (lane == 0) {
        #pragma unroll
        for (int r = 0; r < 8; ++r) out[r] = acc[r];
    } else if (lane == 16) {
        #pragma unroll
        for (int r = 0; r < 8; ++r) out[8 + r] = acc[r];
    }
}

// ---------------------------------------------------------------------------
// Kernel B: reduce K-split partials, apply STATE_DECAY (folded out of the
// matvec), add Gaussian noise, sigmoid -> s_out. Also copies M -> Mout.
// ---------------------------------------------------------------------------
__global__ __launch_bounds__(256) void finalize_s(
    const float* __restrict__ partial, const float* __restrict__ Min,
    float* __restrict__ sOut, float* __restrict__ Mout) {
    int i = blockIdx.x * 256 + threadIdx.x;
    if (i >= NN) return;
    float acc = 0.0f;
    #pragma unroll
    for (int p = 0; p < KSPLIT; ++p) acc += partial[(size_t)p * NN + i];
    acc *= 0.95f;                                   // STATE_DECAY
    unsigned h1 = pcg_hash(0x9E3779B9u ^ (unsigned)i);
    unsigned h2 = pcg_hash(h1 ^ 0x85EBCA6Bu);
    float nz = 0.05f * sqrtf(-2.0f * logf(u01(h1))) *
               cosf(6.28318530718f * u01(h2));      // NOISE_STD * N(0,1)
    sOut[i] = 1.0f / (1.0f + expf(-(acc + nz)));
    Mout[i] = Min[i];
}

// ---------------------------------------------------------------------------
// Kernel C: parallel Gumbel-max categorical sample per spark (reads input W).
// ---------------------------------------------------------------------------
__global__ __launch_bounds__(256) void spark_sample(
    const float* __restrict__ W, const int* __restrict__ sparkPos,
    int* __restrict__ nxtWs, float* __restrict__ posOut) {
    __shared__ float bs[256];
    __shared__ int   bi[256];
    const int i = blockIdx.x;
    const int t = threadIdx.x;
    const int prev = sparkPos[i];
    const float* row = W + (size_t)prev * NN;
    float best = -3.0e38f; int bidx = 0;
    for (int j = t; j < NN; j += 256) {
        float w = fmaxf(row[j], 0.0f) + 1e-6f;
        unsigned h = pcg_hash(((unsigned)i * 131071u) ^ (unsigned)j ^ 0xB5297A4Du);
        float g = -logf(-logf(u01(h)));             // Gumbel(0,1)
        float sc = logf(w) + g;
        if (sc > best || (sc == best && j < bidx)) { best = sc; bidx = j; }
    }
    bs[t] = best; bi[t] = bidx;
    __syncthreads();
    for (int off = 128; off > 0; off >>= 1) {
        if (t < off) {
            if (bs[t + off] > bs[t] ||
                (bs[t + off] == bs[t] && bi[t + off] < bi[t])) {
                bs[t] = bs[t + off]; bi[t] = bi[t + off];
            }
        }
        __syncthreads();
    }
    if (t == 0) { nxtWs[i] = bi[0]; posOut[i] = (float)bi[0]; }
}

// ---------------------------------------------------------------------------
// Kernel D: exact sequential fixup. Runs AFTER the fused decay+clip pass so
// the ordered edge writes land on the clipped Wout (reference order:
// edge-update -> decay -> clip). Handles duplicate-edge chaining, ordered M
// decay, and s[nxt]=1 effects on later s[prev] reads. All 256 threads emit
// the energy output in parallel.
// ---------------------------------------------------------------------------
__global__ __launch_bounds__(256) void spark_scan(
    const float* __restrict__ Win, const int* __restrict__ sparkPos,
    const int* __restrict__ nxtWs, const float* __restrict__ energyIn,
    float* __restrict__ Wout, float* __restrict__ sOut,
    float* __restrict__ Mout, float* __restrict__ energyOut) {
    __shared__ int   shN[KS], shP[KS];
    __shared__ float shV[KS];
    const int t = threadIdx.x;
    shN[t] = nxtWs[t];
    shP[t] = sparkPos[t];
    energyOut[t] = energyIn[t] * 0.98f;
    __syncthreads();
    if (t == 0) {
        for (int i = 0; i < KS; ++i) {
            const int nxt = shN[i], prev = shP[i];
            float cur = Win[(size_t)nxt * NN + prev];
            for (int j = 0; j < i; ++j)                  // latest duplicate wins
                if (shN[j] == nxt && shP[j] == prev) cur = shV[j];
            float v = cur * 0.95f + sOut[prev] * 0.05f;  // LR_EDGE = 0.05
            shV[i] = v;
            Wout[(size_t)nxt * NN + prev] = clip1(v * 0.999f);
            const int gy = nxt >> 7, gx = nxt & (GRIDW - 1);
            const int mi = gy * GRIDW + gx;
            Mout[mi] = Mout[mi] * 0.92f + 0.15f;         // MEM_DECAY, LR_MEMORY
            sOut[nxt] = 1.0f;
        }
    }
}

extern "C" void kernel_launch(void* const* d_in, const int* in_sizes, int n_in,
                              void* d_out, int out_size, void* d_ws, size_t ws_size,
                              hipStream_t stream) {
    (void)in_sizes; (void)n_in; (void)out_size; (void)ws_size;
    const float* W        = (const float*)d_in[0];
    const float* s        = (const float*)d_in[1];
    const float* M        = (const float*)d_in[2];
    const int*   sparkPos = (const int*)d_in[3];
    const float* energy   = (const float*)d_in[4];

    float* out       = (float*)d_out;
    float* Wout      = out;                              // N*N
    float* sOut      = out + (size_t)NN * NN;            // N
    float* Mout      = sOut + NN;                        // 8192
    float* posOut    = Mout + NN;                        // K (as floats)
    float* energyOut = posOut + KS;                      // K

    float* partial = (float*)d_ws;                       // KSPLIT*NN floats
    int*   nxtWs   = (int*)(partial + (size_t)KSPLIT * NN); // K ints

    dim3 gA(NN / 128, KSPLIT);
    matvec_fused<<<gA, 256, 0, stream>>>(W, s, Wout, partial);
    finalize_s  <<<NN / 256, 256, 0, stream>>>(partial, M, sOut, Mout);
    spark_sample<<<KS, 256, 0, stream>>>(W, sparkPos, nxtWs, posOut);
    spark_scan  <<<1, KS, 0, stream>>>(W, sparkPos, nxtWs, energy,
                                       Wout, sOut, Mout, energyOut);
}